// ConvSpatialPropagationNet_71949292143069
// MI455X (gfx1250) — compile-verified
//
#include <hip/hip_runtime.h>

#define TW 256  // pixels per block (one output row segment), 8 wave32 waves

// ---- CDNA5 async global -> LDS staging -------------------------------------
// VGLOBAL encoding, GVS mode: mem_addr = SADDR_u64 + VADDR_i32 + IOFFSET.
// The "dst" operand is a VGPR holding the LDS byte address. Tracked by ASYNCcnt.
__device__ __forceinline__ void async_load_f32_to_lds(const float* sgpr_rowbase,
                                                      int vgpr_byte_off,
                                                      unsigned lds_byte_addr) {
  asm volatile("global_load_async_to_lds_b32 %0, %1, %2 offset:0"
               :: "v"(lds_byte_addr), "v"(vgpr_byte_off), "s"(sgpr_rowbase)
               : "memory");
}

__device__ __forceinline__ void wait_asynccnt0() {
#if __has_builtin(__builtin_amdgcn_s_wait_asynccnt)
  __builtin_amdgcn_s_wait_asynccnt(0);
#else
  asm volatile("s_wait_asynccnt 0" ::: "memory");
#endif
}

// ---------------------------------------------------------------------------
// Fused CSPN kernel: gate normalization + 24-step propagation in registers.
// guidance: (B,8,H,W) f32, blur/sparse: (B,1,H,W) f32, sum_w: (8) f32.
__global__ __launch_bounds__(TW) void cspn_fused_kernel(
    const float* __restrict__ guidance,
    const float* __restrict__ blur,
    const float* __restrict__ sparse,
    const float* __restrict__ sum_w,
    float* __restrict__ out,
    int H, int W)
{
  // Per-channel neighbor offsets derived from _PADS: (dy,dx) = (1-t, 1-l)
  constexpr int DY[8] = {1, 0, -1, 1, -1, 1, 0, -1};
  constexpr int DX[8] = {1, 1, 1, 0, 0, -1, -1, -1};

  __shared__ float gbuf[8][TW + 2];  // 8 channel rows with +-1 column halo

  const int tid = threadIdx.x;
  const int w0  = blockIdx.x * TW;
  const int h   = blockIdx.y;
  const int b   = blockIdx.z;

  // ---- Stage the 8 shifted guidance rows into LDS (async, zero-padded) ----
#pragma unroll
  for (int c = 0; c < 8; ++c) {
    const int row = h + DY[c];
    const bool row_ok = (row >= 0) && (row < H);
    const int rowc = row_ok ? row : 0;  // keep pointer arithmetic in-bounds
    const float* rowbase =
        guidance + (((size_t)b * 8 + c) * (size_t)H + (size_t)rowc) * (size_t)W;
#pragma unroll
    for (int rep = 0; rep < 2; ++rep) {
      const int p = tid + rep * TW;          // covers 0 .. TW+1
      if (p < TW + 2) {
        const int col = w0 - 1 + p;
        if (row_ok && col >= 0 && col < W) {
          async_load_f32_to_lds(rowbase, col * 4,
                                (unsigned)(unsigned long long)&gbuf[c][p]);
        } else {
          gbuf[c][p] = 0.0f;                 // zero-pad outside the image
        }
      }
    }
  }
  wait_asynccnt0();
  __syncthreads();

  const int w = w0 + tid;
  if (w >= W) return;  // tail block; no barriers past this point

  // ---- gate_sum = (sum_c g_c*w_c) / (sum_c |g_c|*w_c or 1) ----------------
  float num = 0.0f, den = 0.0f;
#pragma unroll
  for (int c = 0; c < 8; ++c) {
    const float wc = sum_w[c];                       // uniform (scalar) loads
    const float g  = gbuf[c][tid + 1 + DX[c]];       // stride-1, conflict-free
    num = fmaf(g,        wc, num);
    den = fmaf(fabsf(g), wc, den);
  }
  const float denom = (den == 0.0f) ? 1.0f : den;
  const float gs = num / denom;

  // ---- 24-step propagation (scalar fixed-point recurrence) ----------------
  const size_t pix = ((size_t)b * (size_t)H + (size_t)h) * (size_t)W + (size_t)w;
  const float raw = blur[pix];
  const float sp  = sparse[pix];
  const float m   = (sp > 0.0f) ? 1.0f : ((sp < 0.0f) ? -1.0f : 0.0f);  // sign
  const float base = (1.0f - gs) * raw;  // (1 - gate_sum) * raw, invariant
  float d = raw;
#pragma unroll
  for (int it = 0; it < 24; ++it) {
    const float dn = fmaf(gs, d, base);        // (1-gs)*raw + nws, nws = gs*d
    d = fmaf(1.0f - m, dn, m * raw);           // (1-mask)*d + mask*raw
  }
  out[pix] = d;
}

extern "C" void kernel_launch(void* const* d_in, const int* in_sizes, int n_in,
                              void* d_out, int out_size, void* d_ws, size_t ws_size,
                              hipStream_t stream) {
  (void)in_sizes; (void)n_in; (void)d_ws; (void)ws_size; (void)out_size;
  const float* guidance = (const float*)d_in[0];
  const float* blur     = (const float*)d_in[1];
  const float* sparse   = (const float*)d_in[2];
  const float* sumw     = (const float*)d_in[3];
  float* out            = (float*)d_out;

  const int B = 8, H = 352, W = 1216;
  dim3 grid((W + TW - 1) / TW, H, B);
  cspn_fused_kernel<<<grid, dim3(TW), 0, stream>>>(guidance, blur, sparse, sumw,
                                                   out, H, W);
}